// SelfAttention_43817256354353
// MI455X (gfx1250) — compile-verified
//
#include <hip/hip_runtime.h>
#include <math.h>

// Problem constants (match reference)
#define B_ 4
#define L_ 2048
#define D_ 256          // model dim == head dim
#define H_ 8
#define DH (D_ * H_)    // 2048
#define M_ (B_ * L_)    // 8192

typedef __attribute__((ext_vector_type(2))) float v2f;
typedef __attribute__((ext_vector_type(8))) float v8f;
typedef __attribute__((ext_vector_type(4))) int v4i;

typedef __attribute__((address_space(1))) v4i g_v4i;  // global int4
typedef __attribute__((address_space(3))) v4i l_v4i;  // LDS int4

// CDNA5 async global->LDS DMA path (ASYNCcnt-tracked). Guarded so the file
// compiles on toolchains without the builtin (falls back to VGPR staging).
#if defined(__has_builtin)
#if __has_builtin(__builtin_amdgcn_global_load_async_to_lds_b128) && \
    __has_builtin(__builtin_amdgcn_s_wait_asynccnt)
#define USE_ASYNC_LDS 1
#endif
#endif
#ifndef USE_ASYNC_LDS
#define USE_ASYNC_LDS 0
#endif

static __device__ __forceinline__ void copy16_to_lds(const float* g, float* l) {
#if USE_ASYNC_LDS
  __builtin_amdgcn_global_load_async_to_lds_b128((g_v4i*)g, (l_v4i*)l, 0, 0);
#else
  *(float4*)l = *(const float4*)g;
#endif
}
static __device__ __forceinline__ void lds_copy_join() {
#if USE_ASYNC_LDS
  __builtin_amdgcn_s_wait_asynccnt(0);
#endif
}

// D = A(16x4 f32) * B(4x16 f32) + C(16x16 f32)  — CDNA5 V_WMMA_F32_16X16X4_F32
static __device__ __forceinline__ v8f wmma_f32(v2f a, v2f b, v8f c) {
  return __builtin_amdgcn_wmma_f32_16x16x4_f32(
      /*neg_a=*/false, a, /*neg_b=*/false, b,
      /*c_mod=*/(short)0, c, /*reuse_a=*/false, /*reuse_b=*/false);
}

// ---------------------------------------------------------------------------
// Kernel 1: QKV projections.  C[M, DH] = x[M, D] @ W[D, DH], scattered into
// [B, H, L, D] layout. blockIdx.z selects q/k/v. 64x64 tile, 8 waves.
// ---------------------------------------------------------------------------
__global__ __launch_bounds__(256) void qkv_proj_kernel(
    const float* __restrict__ x, const float* __restrict__ Wq,
    const float* __restrict__ Wk, const float* __restrict__ Wv,
    float* __restrict__ Qb, float* __restrict__ Kb, float* __restrict__ Vb) {
  const int which = blockIdx.z;
  const float* W = (which == 0) ? Wq : (which == 1) ? Wk : Wv;
  float* Out = (which == 0) ? Qb : (which == 1) ? Kb : Vb;

  const int m0 = blockIdx.x * 64;
  const int n0 = blockIdx.y * 64;

  __shared__ float As[64][20];  // stride 20: banks 20r%64 distinct, 16B rows
  __shared__ float Bs[16][68];  // stride 68: 16B-aligned rows

  const int tid = threadIdx.x;
  const int lane = tid & 31;
  const int wave = tid >> 5;    // 0..7
  const int wm = wave >> 1;     // 0..3  (16-row strip)
  const int wn = wave & 1;      // 0..1  (32-col strip)
  const int row16 = lane & 15;
  const int half = lane >> 4;
  const int kb = half * 2;

  v8f acc[2];
  v8f zero = {};
  acc[0] = zero; acc[1] = zero;

  const int ar = tid >> 2, ac = (tid & 3) * 4;
  const int br = tid >> 4, bc = (tid & 15) * 4;

  for (int k0 = 0; k0 < D_; k0 += 16) {
    __syncthreads();
    copy16_to_lds(x + (size_t)(m0 + ar) * D_ + k0 + ac, &As[ar][ac]);
    copy16_to_lds(W + (size_t)(k0 + br) * DH + n0 + bc, &Bs[br][bc]);
    lds_copy_join();
    __syncthreads();
#pragma unroll
    for (int kk = 0; kk < 16; kk += 4) {
      v2f a;
      a.x = As[wm * 16 + row16][kk + kb];
      a.y = As[wm * 16 + row16][kk + kb + 1];
#pragma unroll
      for (int t = 0; t < 2; ++t) {
        v2f b;
        b.x = Bs[kk + kb][wn * 32 + t * 16 + row16];
        b.y = Bs[kk + kb + 1][wn * 32 + t * 16 + row16];
        acc[t] = wmma_f32(a, b, acc[t]);
      }
    }
  }

  // scatter: col n -> (h = n>>8, d = n&255); row m -> (b = m>>11, l = m&2047)
#pragma unroll
  for (int t = 0; t < 2; ++t) {
    const int ng = n0 + wn * 32 + t * 16 + row16;
    const int hh = ng >> 8, dd = ng & 255;
#pragma unroll
    for (int r = 0; r < 8; ++r) {
      const int mg = m0 + wm * 16 + r + half * 8;
      const int bb = mg >> 11, ll = mg & 2047;
      Out[((size_t)((bb * H_ + hh) * L_) + ll) * D_ + dd] = acc[t][r];
    }
  }
}

// ---------------------------------------------------------------------------
// Kernel 2: flash attention. One block per (b, h, 64 q-rows); 4 waves x 16
// q-rows. K/V tiles of 64x256 streamed to LDS via async DMA; streaming
// softmax with per-row running max/sum.
// ---------------------------------------------------------------------------
__global__ __launch_bounds__(128) void attn_kernel(
    const float* __restrict__ Qb, const float* __restrict__ Kb,
    const float* __restrict__ Vb, const int* __restrict__ mask,
    float* __restrict__ Ob) {
  const int q0 = blockIdx.x * 64;
  const int h = blockIdx.y;
  const int b = blockIdx.z;

  __shared__ float Ks[64][260];   // pad 4: S b-frag banks 4r%64, conflict-free
  __shared__ float Vs[64][264];   // pad 8: half1 bank shift +16 -> disjoint
  __shared__ float Sw[4][16][68]; // per-wave 16x64 score strip (pad 4)

  const int tid = threadIdx.x;
  const int lane = tid & 31;
  const int wave = tid >> 5;           // 0..3
  const int row16 = lane & 15;
  const int half = lane >> 4;
  const int kb = half * 2;

  const size_t head_off = (size_t)(b * H_ + h) * L_ * D_;
  const float* Qh = Qb + head_off;
  const float* Kh = Kb + head_off;
  const float* Vh = Vb + head_off;

  // Preload this wave's Q A-fragments (16 rows x 256 feats) into registers.
  v2f qa[64];
  {
    const float* qrow = Qh + (size_t)(q0 + wave * 16 + row16) * D_;
#pragma unroll
    for (int i = 0; i < 64; ++i) {
      const float2 t2 = *(const float2*)(qrow + i * 4 + kb);
      qa[i].x = t2.x; qa[i].y = t2.y;
    }
  }

  v8f zero = {};
  v8f oacc[16];
#pragma unroll
  for (int n = 0; n < 16; ++n) oacc[n] = zero;

  float m_run = -INFINITY;
  float l_run = 0.0f;
  const float scale = 0.0625f;  // 1/sqrt(256)

  const int ldr = tid >> 1;            // K/V tile row this thread fills
  const int ldc = (tid & 1) * 128;     // 128-column half

  for (int k0 = 0; k0 < L_; k0 += 64) {
    __syncthreads();  // previous iteration done with Ks/Vs/Sw
    {
      const float* kr = Kh + (size_t)(k0 + ldr) * D_ + ldc;
      const float* vr = Vh + (size_t)(k0 + ldr) * D_ + ldc;
#pragma unroll
      for (int i = 0; i < 32; ++i) {
        copy16_to_lds(kr + i * 4, &Ks[ldr][ldc + i * 4]);
        copy16_to_lds(vr + i * 4, &Vs[ldr][ldc + i * 4]);
      }
      lds_copy_join();
    }
    __syncthreads();

    // S strip (16 x 64) = Q(16x256) @ K^T(256x64)
    v8f sacc[4];
#pragma unroll
    for (int t = 0; t < 4; ++t) sacc[t] = zero;
#pragma unroll
    for (int k = 0; k < 64; ++k) {
      const v2f a = qa[k];
#pragma unroll
      for (int t = 0; t < 4; ++t) {
        v2f bf;
        bf.x = Ks[t * 16 + row16][4 * k + kb];
        bf.y = Ks[t * 16 + row16][4 * k + kb + 1];
        sacc[t] = wmma_f32(a, bf, sacc[t]);
      }
    }
#pragma unroll
    for (int t = 0; t < 4; ++t)
#pragma unroll
      for (int r = 0; r < 8; ++r)
        Sw[wave][r + half * 8][t * 16 + row16] = sacc[t][r];
    __syncthreads();

    // Softmax phase 1: scale + mask (int4 loads) + running row max.
    // Lane handles row=row16, 32-column half selected by `half`.
    const int qg = q0 + wave * 16 + row16;
    const int* mrow = mask + ((size_t)b * L_ + qg) * L_ + k0 + half * 32;
    float lmax = -INFINITY;
#pragma unroll
    for (int i = 0; i < 8; ++i) {
      const int4 m4 = *(const int4*)(mrow + i * 4);
      const int cc = half * 32 + i * 4;
      float s0 = Sw[wave][row16][cc + 0] * scale;
      float s1 = Sw[wave][row16][cc + 1] * scale;
      float s2 = Sw[wave][row16][cc + 2] * scale;
      float s3 = Sw[wave][row16][cc + 3] * scale;
      s0 = (m4.x == 0) ? -INFINITY : s0;
      s1 = (m4.y == 0) ? -INFINITY : s1;
      s2 = (m4.z == 0) ? -INFINITY : s2;
      s3 = (m4.w == 0) ? -INFINITY : s3;
      Sw[wave][row16][cc + 0] = s0;
      Sw[wave][row16][cc + 1] = s1;
      Sw[wave][row16][cc + 2] = s2;
      Sw[wave][row16][cc + 3] = s3;
      lmax = fmaxf(lmax, fmaxf(fmaxf(s0, s1), fmaxf(s2, s3)));
    }
    lmax = fmaxf(lmax, __shfl_xor(lmax, 16, 32));
    const float m_new = fmaxf(m_run, lmax);
    const float alpha = (m_run == -INFINITY) ? 0.0f : __expf(m_run - m_new);

    // Phase 2: exponentiate + row sum
    float lsum = 0.0f;
#pragma unroll 8
    for (int c = 0; c < 32; ++c) {
      const int cc = half * 32 + c;
      const float p = __expf(Sw[wave][row16][cc] - m_new);
      Sw[wave][row16][cc] = p;
      lsum += p;
    }
    lsum += __shfl_xor(lsum, 16, 32);
    l_run = l_run * alpha + lsum;
    m_run = m_new;
    __syncthreads();

    // Rescale O by per-row alpha (lane r holds alpha for row r)
    float arow[8];
#pragma unroll
    for (int r = 0; r < 8; ++r) arow[r] = __shfl(alpha, r + half * 8, 32);
#pragma unroll
    for (int n = 0; n < 16; ++n)
#pragma unroll
      for (int r = 0; r < 8; ++r) oacc[n][r] *= arow[r];

    // O += P(16x64) @ V(64x256)
#pragma unroll
    for (int k = 0; k < 16; ++k) {
      v2f a;
      a.x = Sw[wave][row16][4 * k + kb];
      a.y = Sw[wave][row16][4 * k + kb + 1];
#pragma unroll
      for (int n = 0; n < 16; ++n) {
        v2f bf;
        bf.x = Vs[4 * k + kb][n * 16 + row16];
        bf.y = Vs[4 * k + kb + 1][n * 16 + row16];
        oacc[n] = wmma_f32(a, bf, oacc[n]);
      }
    }
  }

  // Finalize: O /= l, write [B, L, H*D] (head-major cols) for output GEMM
  float lrow[8];
#pragma unroll
  for (int r = 0; r < 8; ++r)
    lrow[r] = 1.0f / __shfl(l_run, r + half * 8, 32);
#pragma unroll
  for (int n = 0; n < 16; ++n)
#pragma unroll
    for (int r = 0; r < 8; ++r) {
      const int qg2 = q0 + wave * 16 + r + half * 8;
      Ob[(size_t)(b * L_ + qg2) * DH + h * D_ + n * 16 + row16] =
          oacc[n][r] * lrow[r];
    }
}

// ---------------------------------------------------------------------------
// Kernel 3: out = O[M, DH] @ Wu[DH, D] + bu.  Same 64x64 tiling, K = 2048.
// ---------------------------------------------------------------------------
__global__ __launch_bounds__(256) void out_proj_kernel(
    const float* __restrict__ A, const float* __restrict__ Wu,
    const float* __restrict__ bu, float* __restrict__ out) {
  const int m0 = blockIdx.x * 64;
  const int n0 = blockIdx.y * 64;

  __shared__ float As[64][20];
  __shared__ float Bs[16][68];

  const int tid = threadIdx.x;
  const int lane = tid & 31;
  const int wave = tid >> 5;
  const int wm = wave >> 1;
  const int wn = wave & 1;
  const int row16 = lane & 15;
  const int half = lane >> 4;
  const int kb = half * 2;

  v8f acc[2];
  v8f zero = {};
  acc[0] = zero; acc[1] = zero;

  const int ar = tid >> 2, ac = (tid & 3) * 4;
  const int br = tid >> 4, bc = (tid & 15) * 4;

  for (int k0 = 0; k0 < DH; k0 += 16) {
    __syncthreads();
    copy16_to_lds(A + (size_t)(m0 + ar) * DH + k0 + ac, &As[ar][ac]);
    copy16_to_lds(Wu + (size_t)(k0 + br) * D_ + n0 + bc, &Bs[br][bc]);
    lds_copy_join();
    __syncthreads();
#pragma unroll
    for (int kk = 0; kk < 16; kk += 4) {
      v2f a;
      a.x = As[wm * 16 + row16][kk + kb];
      a.y = As[wm * 16 + row16][kk + kb + 1];
#pragma unroll
      for (int t = 0; t < 2; ++t) {
        v2f b;
        b.x = Bs[kk + kb][wn * 32 + t * 16 + row16];
        b.y = Bs[kk + kb + 1][wn * 32 + t * 16 + row16];
        acc[t] = wmma_f32(a, b, acc[t]);
      }
    }
  }

#pragma unroll
  for (int t = 0; t < 2; ++t) {
    const int ng = n0 + wn * 32 + t * 16 + row16;
    const float bias = bu[ng];
#pragma unroll
    for (int r = 0; r < 8; ++r) {
      const int mg = m0 + wm * 16 + r + half * 8;
      out[(size_t)mg * D_ + ng] = acc[t][r] + bias;
    }
  }
}

// ---------------------------------------------------------------------------
extern "C" void kernel_launch(void* const* d_in, const int* in_sizes, int n_in,
                              void* d_out, int out_size, void* d_ws,
                              size_t ws_size, hipStream_t stream) {
  (void)in_sizes; (void)n_in; (void)out_size; (void)ws_size;
  const float* x   = (const float*)d_in[0];
  const int*  mask = (const int*)d_in[1];
  const float* Wq  = (const float*)d_in[2];
  const float* Wk  = (const float*)d_in[3];
  const float* Wv  = (const float*)d_in[4];
  const float* Wu  = (const float*)d_in[5];
  const float* bu  = (const float*)d_in[6];
  float* out = (float*)d_out;

  const size_t SZ = (size_t)B_ * H_ * L_ * D_;  // 16.78M floats per buffer
  float* Qb = (float*)d_ws;
  float* Kb = Qb + SZ;
  float* Vb = Kb + SZ;
  float* Ob = Vb + SZ;

  qkv_proj_kernel<<<dim3(M_ / 64, DH / 64, 3), 256, 0, stream>>>(
      x, Wq, Wk, Wv, Qb, Kb, Vb);
  attn_kernel<<<dim3(L_ / 64, H_, B_), 128, 0, stream>>>(
      Qb, Kb, Vb, mask, Ob);
  out_proj_kernel<<<dim3(M_ / 64, D_ / 64), 256, 0, stream>>>(
      Ob, Wu, bu, out);
}